// dec_45827301048329
// MI455X (gfx1250) — compile-verified
//
#include <hip/hip_runtime.h>
#include <hip/hip_bf16.h>

// Dims from the reference
#define DV 50257
#define DE 300
#define DH 300
#define DC 600
#define DS 8192
#define DL 1024
#define D3H 900
#define KP 150            // K pairs: 300/2
#define N1P 912           // 900 padded to 57 tiles
#define N2P 304           // 300 padded to 19 tiles

typedef __attribute__((ext_vector_type(2))) float v2f;
typedef __attribute__((ext_vector_type(8))) float v8f;

__device__ __forceinline__ float sigmoidf_(float x) { return 1.0f / (1.0f + __expf(-x)); }
__device__ __forceinline__ float leaky_(float x) { return x >= 0.0f ? x : 0.01f * x; }

// ---------------------------------------------------------------------------
// Setup: zero per-step accumulators, gather embeddings into WMMA-packed
// layout, pack weight e-column blocks (zero-padded) into WMMA layout,
// copy initial hidden states.
//
// Packed layouts (float2 granularity, p = K-pair index 0..149):
//   E_pack [p][m]  m in 0..1023       -> float2 (E[m][2p], E[m][2p+1])
//   B1p    [p][n]  n in 0..911 (pad0) -> float2 (W_ih1[n][2p], W_ih1[n][2p+1])
//   B2p    [p][n]  n in 0..303 (pad0) -> float2 (W_pre[n][2p], W_pre[n][2p+1])
// ---------------------------------------------------------------------------
__global__ void k_setup(const int* __restrict__ inputs, const float* __restrict__ emb,
                        const float* __restrict__ h1_in, const float* __restrict__ h2_in,
                        const float* __restrict__ W_ih1, const float* __restrict__ W_pre,
                        float* __restrict__ sums, float* __restrict__ ctxAcc,
                        float* __restrict__ E_pack, float* __restrict__ B1p,
                        float* __restrict__ B2p, float* __restrict__ g_h1,
                        float* __restrict__ g_h2) {
    const int NZ  = DL + DL * DC;       // sums + ctxAcc zero
    const int NE  = DL * DE;            // embedding gather
    const int NB1 = KP * N1P * 2;       // W_ih1 e-block pack
    const int NB2 = KP * N2P * 2;       // W_pre e-block pack
    int idx = blockIdx.x * blockDim.x + threadIdx.x;
    if (idx < NZ) {
        if (idx < DL) sums[idx] = 0.0f;
        else ctxAcc[idx - DL] = 0.0f;
    } else if (idx < NZ + NE) {
        int j = idx - NZ;
        int t = j / DE;
        int k = j - t * DE;
        int tok = (t == 0) ? 0 : inputs[t - 1];
        E_pack[((k >> 1) * DL + t) * 2 + (k & 1)] = emb[tok * DE + k];
    } else if (idx < NZ + NE + NB1) {
        int j = idx - NZ - NE;
        int c = j & 1;
        int pn = j >> 1;
        int p = pn / N1P;
        int n = pn - p * N1P;
        B1p[j] = (n < D3H) ? W_ih1[n * (DE + DC) + 2 * p + c] : 0.0f;
    } else if (idx < NZ + NE + NB1 + NB2) {
        int j = idx - NZ - NE - NB1;
        int c = j & 1;
        int pn = j >> 1;
        int p = pn / N2P;
        int n = pn - p * N2P;
        B2p[j] = (n < DH) ? W_pre[n * (DE + DH + DC) + 2 * p + c] : 0.0f;
    } else if (idx < NZ + NE + NB1 + NB2 + DH) {
        int i = idx - NZ - NE - NB1 - NB2;
        g_h1[i] = h1_in[i];
    } else if (idx < NZ + NE + NB1 + NB2 + 2 * DH) {
        int i = idx - NZ - NE - NB1 - NB2 - DH;
        g_h2[i] = h2_in[i];
    }
}

// Initial query: qb = Wq @ h1 + bq
__global__ void k_q0(const float* __restrict__ Wq, const float* __restrict__ bq,
                     const float* __restrict__ h1_in, float* __restrict__ qb) {
    int i = threadIdx.x;
    if (i < DH) {
        const float* wr = Wq + i * DH;
        float acc = bq[i];
        for (int k = 0; k < DH; ++k) acc += wr[k] * h1_in[k];
        qb[i] = acc;
    }
}

// ---------------------------------------------------------------------------
// WMMA fp32 GEMM on pre-packed operands.
// D[1024 x Npad] = E @ Wt ; one 16x16 tile per wave, K stepped by 4 via
// V_WMMA_F32_16X16X4_F32.  A/B loads are unconditional coalesced b64 from
// packed layout; EXEC stays all-ones (guard branch is wave-uniform).
// ---------------------------------------------------------------------------
__global__ void k_gemm_wmma(const float* __restrict__ Ap, const float* __restrict__ Bp,
                            float* __restrict__ D, int Ntiles) {
    int wid  = threadIdx.x >> 5;
    int lane = threadIdx.x & 31;
    int wglobal = blockIdx.x * (blockDim.x >> 5) + wid;
    int total = (DL / 16) * Ntiles;
    if (wglobal >= total) return;                 // wave-uniform
    int tm = wglobal / Ntiles;
    int tn = wglobal - tm * Ntiles;

    int half = lane >> 4;                         // lanes 0-15: K 2p,2p+1 of k4; 16-31: +2
    int m = tm * 16 + (lane & 15);
    int n = tn * 16 + (lane & 15);
    int Npad = Ntiles * 16;

    const v2f* pa = (const v2f*)Ap + half * DL + m;
    const v2f* pb = (const v2f*)Bp + half * Npad + n;

    v8f acc = {0.0f, 0.0f, 0.0f, 0.0f, 0.0f, 0.0f, 0.0f, 0.0f};
#pragma unroll 5
    for (int it = 0; it < KP / 2; ++it) {         // 75 iterations (K=300, 4/step)
        v2f a = *pa;
        v2f b = *pb;
        acc = __builtin_amdgcn_wmma_f32_16x16x4_f32(false, a, false, b,
                                                    (short)0, acc, false, false);
        pa += 2 * DL;                             // advance 2 K-pairs
        pb += 2 * Npad;
    }
    // C/D layout: VGPR g -> row tm*16 + g (+8 for lanes 16..31), col = lane&15
    int row0 = tm * 16 + (half ? 8 : 0);
    int col  = tn * 16 + (lane & 15);
#pragma unroll
    for (int g = 0; g < 8; ++g) D[(row0 + g) * Npad + col] = acc[g];
}

// ---------------------------------------------------------------------------
// Attention scores: w[s] = exp( v . tanh(PK[s,:] + qb) ); block partial sums
// of w accumulated into sums[t] with f32 atomics.  One wave per key row.
// ---------------------------------------------------------------------------
__global__ void k_scores(const float* __restrict__ PK, const float* __restrict__ vattn,
                         const float* __restrict__ qb, float* __restrict__ wbuf,
                         float* __restrict__ sums, int t) {
    __shared__ float part[8];
    int wid  = threadIdx.x >> 5;
    int lane = threadIdx.x & 31;
    int s = blockIdx.x * 8 + wid;                 // 1024 blocks * 8 waves = 8192 rows
    const float* row = PK + s * DH;
    float acc = 0.0f;
    for (int k = lane; k < DH; k += 32) {
        __builtin_prefetch(row + k + 256, 0, 0);  // global_prefetch_b8
        acc += vattn[k] * tanhf(row[k] + qb[k]);
    }
#pragma unroll
    for (int off = 16; off > 0; off >>= 1) acc += __shfl_down(acc, off, 32);
    if (lane == 0) {
        float e = __expf(acc);                    // scores bounded by sum|v| ~ 12
        wbuf[s] = e;
        part[wid] = e;
    }
    __syncthreads();
    if (threadIdx.x == 0) {
        float bs = 0.0f;
#pragma unroll
        for (int i = 0; i < 8; ++i) bs += part[i];
        atomicAdd(&sums[t], bs);
    }
}

// ---------------------------------------------------------------------------
// Context: ctxAcc[t, c] += sum_s w[s] * EO[s, c] (64 rows per block, coalesced
// over columns), and normalized alphas written straight to d_out.
// ---------------------------------------------------------------------------
__global__ void k_ctx(const float* __restrict__ EO, const float* __restrict__ wbuf,
                      const float* __restrict__ sums, float* __restrict__ ctxAcc,
                      float* __restrict__ alphas, int t) {
    __shared__ float sw[64];
    int s0 = blockIdx.x * 64;
    float inv = 1.0f / sums[t];
    if (threadIdx.x < 64) {
        float wv = wbuf[s0 + threadIdx.x];
        sw[threadIdx.x] = wv;
        alphas[(size_t)t * DS + s0 + threadIdx.x] = wv * inv;
    }
    __syncthreads();
    for (int c = threadIdx.x; c < DC; c += 256) {
        float acc = 0.0f;
#pragma unroll 8
        for (int r = 0; r < 64; ++r) acc += sw[r] * EO[(s0 + r) * DC + c];
        atomicAdd(&ctxAcc[t * DC + c], acc);
    }
}

// ---------------------------------------------------------------------------
// Fused recurrent cell: GRU1 -> GRU2 -> pre-projection -> next query.
// Single 1024-thread workgroup (32 wave32 waves); intermediates in LDS.
// ---------------------------------------------------------------------------
__global__ void k_cell(const float* __restrict__ GiE1, const float* __restrict__ PreE,
                       const float* __restrict__ ctxAcc, const float* __restrict__ sums,
                       const float* __restrict__ W_ih1, const float* __restrict__ W_hh1,
                       const float* __restrict__ b_ih1, const float* __restrict__ b_hh1,
                       const float* __restrict__ W_ih2, const float* __restrict__ W_hh2,
                       const float* __restrict__ b_ih2, const float* __restrict__ b_hh2,
                       const float* __restrict__ W_pre, const float* __restrict__ b_pre,
                       const float* __restrict__ Wq, const float* __restrict__ bq,
                       float* __restrict__ g_h1, float* __restrict__ g_h2,
                       float* __restrict__ qb, float* __restrict__ outs,
                       float* __restrict__ pres, float* __restrict__ h1f,
                       float* __restrict__ h2f, int t) {
    __shared__ float sh1[DH], sh2[DH], sctx[DC], sgi[D3H], sgh[D3H];
    __shared__ float so1[DH], sh1n[DH], so2[DH];
    int tid = threadIdx.x;

    if (tid < DH) { sh1[tid] = g_h1[tid]; sh2[tid] = g_h2[tid]; }
    if (tid < DC) sctx[tid] = ctxAcc[t * DC + tid] * (1.0f / sums[t]);
    __syncthreads();

    // --- GRU1 gates ---
    if (tid < D3H) {
        const float* wc = W_ih1 + tid * (DE + DC) + DE;   // ctx part of x1
        float a = GiE1[t * N1P + tid] + b_ih1[tid];       // precomputed emb part
        for (int c = 0; c < DC; ++c) a += wc[c] * sctx[c];
        const float* wh = W_hh1 + tid * DH;
        float g = b_hh1[tid];
        for (int k = 0; k < DH; ++k) g += wh[k] * sh1[k];
        sgi[tid] = a; sgh[tid] = g;
    }
    __syncthreads();
    if (tid < DH) {
        float r = sigmoidf_(sgi[tid] + sgh[tid]);
        float z = sigmoidf_(sgi[tid + DH] + sgh[tid + DH]);
        float n = tanhf(sgi[tid + 2 * DH] + r * sgh[tid + 2 * DH]);
        float h1n = (1.0f - z) * n + z * sh1[tid];
        sh1n[tid] = h1n;
        g_h1[tid] = h1n;
        so1[tid] = leaky_(h1n);
    }
    __syncthreads();

    // --- GRU2 gates ---
    if (tid < D3H) {
        const float* w2 = W_ih2 + tid * (DH + DC);
        float a = b_ih2[tid];
        for (int k = 0; k < DH; ++k) a += w2[k] * so1[k];
        for (int c = 0; c < DC; ++c) a += w2[DH + c] * sctx[c];
        const float* wh = W_hh2 + tid * DH;
        float g = b_hh2[tid];
        for (int k = 0; k < DH; ++k) g += wh[k] * sh2[k];
        sgi[tid] = a; sgh[tid] = g;
    }
    __syncthreads();
    if (tid < DH) {
        float r = sigmoidf_(sgi[tid] + sgh[tid]);
        float z = sigmoidf_(sgi[tid + DH] + sgh[tid + DH]);
        float n = tanhf(sgi[tid + 2 * DH] + r * sgh[tid + 2 * DH]);
        float h2n = (1.0f - z) * n + z * sh2[tid];
        g_h2[tid] = h2n;
        float o2 = leaky_(h2n);
        so2[tid] = o2;
        outs[t * DH + tid] = o2;
        if (t == DL - 1) { h1f[tid] = sh1n[tid]; h2f[tid] = h2n; }
    }
    __syncthreads();

    // --- pre-projection + next query ---
    if (tid < DH) {
        const float* wp = W_pre + tid * (DE + DH + DC);
        float p = PreE[t * N2P + tid] + b_pre[tid];       // precomputed emb part
        for (int k = 0; k < DH; ++k) p += wp[DE + k] * so2[k];
        for (int c = 0; c < DC; ++c) p += wp[DE + DH + c] * sctx[c];
        pres[t * DH + tid] = leaky_(p);

        const float* wq = Wq + tid * DH;
        float q = bq[tid];
        for (int k = 0; k < DH; ++k) q += wq[k] * sh1n[k];
        qb[tid] = q;                                      // query for step t+1
    }
}

// ---------------------------------------------------------------------------
extern "C" void kernel_launch(void* const* d_in, const int* in_sizes, int n_in,
                              void* d_out, int out_size, void* d_ws, size_t ws_size,
                              hipStream_t stream) {
    const int*   inputs = (const int*)d_in[0];
    const float* h1_in  = (const float*)d_in[1];
    const float* h2_in  = (const float*)d_in[2];
    const float* PK     = (const float*)d_in[3];
    const float* EO     = (const float*)d_in[4];
    const float* emb    = (const float*)d_in[5];
    const float* Wq     = (const float*)d_in[6];
    const float* bq     = (const float*)d_in[7];
    const float* vattn  = (const float*)d_in[8];
    const float* W_ih1  = (const float*)d_in[9];
    const float* W_hh1  = (const float*)d_in[10];
    const float* b_ih1  = (const float*)d_in[11];
    const float* b_hh1  = (const float*)d_in[12];
    const float* W_ih2  = (const float*)d_in[13];
    const float* W_hh2  = (const float*)d_in[14];
    const float* b_ih2  = (const float*)d_in[15];
    const float* b_hh2  = (const float*)d_in[16];
    const float* W_pre  = (const float*)d_in[17];
    const float* b_pre  = (const float*)d_in[18];

    // Workspace layout (floats)
    float* ws     = (float*)d_ws;
    float* E_pack = ws;                       // 150*1024*2 = 307200
    float* B1p    = E_pack + KP * DL * 2;     // 150*912*2  = 273600
    float* B2p    = B1p    + KP * N1P * 2;    // 150*304*2  =  91200
    float* GiE1   = B2p    + KP * N2P * 2;    // 1024*912   = 933888
    float* PreE   = GiE1   + DL * N1P;        // 1024*304   = 311296
    float* ctxAcc = PreE   + DL * N2P;        // 1024*600   = 614400
    float* sums   = ctxAcc + DL * DC;         // 1024
    float* wbuf   = sums   + DL;              // 8192
    float* qb     = wbuf   + DS;              // 300
    float* g_h1   = qb     + DH;              // 300
    float* g_h2   = g_h1   + DH;              // 300

    // Output layout: outs[1,L,H], pres[1,L,H], alphas[L,S], h1f[H], h2f[H]
    float* out    = (float*)d_out;
    float* outs   = out;
    float* pres   = out + DL * DH;
    float* alphas = out + 2 * DL * DH;
    float* h1f    = alphas + (size_t)DL * DS;
    float* h2f    = h1f + DH;

    // Phase 0: setup/pack + hoisted WMMA GEMMs for embedding-dependent terms
    {
        int n = (DL + DL * DC) + DL * DE + KP * N1P * 2 + KP * N2P * 2 + 2 * DH;
        int nb = (n + 255) / 256;
        k_setup<<<nb, 256, 0, stream>>>(inputs, emb, h1_in, h2_in, W_ih1, W_pre,
                                        sums, ctxAcc, E_pack, B1p, B2p, g_h1, g_h2);
    }
    k_q0<<<1, 320, 0, stream>>>(Wq, bq, h1_in, qb);
    // GiE1[t, j] = sum_k E[t,k] * W_ih1[j, k]   (M=1024, N=900->912, K=300)
    k_gemm_wmma<<<(64 * 57) / 4, 128, 0, stream>>>(E_pack, B1p, GiE1, 57);
    // PreE[t, i] = sum_k E[t,k] * W_pre[i, k]   (M=1024, N=300->304, K=300)
    k_gemm_wmma<<<(64 * 19) / 4, 128, 0, stream>>>(E_pack, B2p, PreE, 19);

    // Sequential decode loop: 3 kernels per step
    for (int t = 0; t < DL; ++t) {
        k_scores<<<DS / 8, 256, 0, stream>>>(PK, vattn, qb, wbuf, sums, t);
        k_ctx<<<DS / 64, 256, 0, stream>>>(EO, wbuf, sums, ctxAcc, alphas, t);
        k_cell<<<1, 1024, 0, stream>>>(GiE1, PreE, ctxAcc, sums,
                                       W_ih1, W_hh1, b_ih1, b_hh1,
                                       W_ih2, W_hh2, b_ih2, b_hh2,
                                       W_pre, b_pre, Wq, bq,
                                       g_h1, g_h2, qb, outs, pres, h1f, h2f, t);
    }
}